// RGCN_12068858101923
// MI455X (gfx1250) — compile-verified
//
#include <hip/hip_runtime.h>

typedef __attribute__((ext_vector_type(2))) float v2f;
typedef __attribute__((ext_vector_type(8))) float v8f;

// ---------------------------------------------------------------- utilities
__global__ __launch_bounds__(256) void zero_u32_kernel(unsigned* __restrict__ p, size_t n) {
  size_t i = (size_t)blockIdx.x * blockDim.x + threadIdx.x;
  size_t stride = (size_t)gridDim.x * blockDim.x;
  for (; i < n; i += stride) p[i] = 0u;
}

__global__ __launch_bounds__(256) void relu_kernel(float* __restrict__ h, size_t n) {
  size_t i = (size_t)blockIdx.x * blockDim.x + threadIdx.x;
  size_t stride = (size_t)gridDim.x * blockDim.x;
  for (; i < n; i += stride) h[i] = fmaxf(h[i], 0.0f);
}

__global__ __launch_bounds__(256) void bias_init_kernel(float* __restrict__ hout,
                                                        const float* __restrict__ b,
                                                        int N, int H) {
  size_t i = (size_t)blockIdx.x * blockDim.x + threadIdx.x;
  if (i < (size_t)N * H) hout[i] = b[i % H];
}

// pack rows (2p, 2p+1) of a (Ktot x H) matrix into float2 pairs:
// Bp[p*H + col] = (B[2p][col], B[2p+1][col])
__global__ __launch_bounds__(256) void pack_pairs_kernel(const float* __restrict__ B,
                                                         float2* __restrict__ Bp,
                                                         int Ktot, int H) {
  int t = blockIdx.x * blockDim.x + threadIdx.x;
  if (t >= (Ktot / 2) * H) return;
  int p = t / H, col = t % H;
  Bp[t] = make_float2(B[(size_t)(2 * p) * H + col], B[(size_t)(2 * p + 1) * H + col]);
}

// ---------------------------------------------------------------- CSR build
// segment id s = etype*N + dst;  edge_index flat: [0..E) = src, [E..2E) = dst
__global__ __launch_bounds__(256) void hist_kernel(const int* __restrict__ ei,
                                                   const int* __restrict__ et,
                                                   unsigned* __restrict__ cnt,
                                                   int E, int N) {
  int e = blockIdx.x * blockDim.x + threadIdx.x;
  if (e >= E) return;
  int s = et[e] * N + ei[(size_t)E + e];
  atomicAdd(&cnt[s], 1u);
}

// one workgroup: exclusive prefix-sum of cnt[0..n) -> rowstart[0..n], plus inv_cnt
__global__ __launch_bounds__(1024) void scan_kernel(const unsigned* __restrict__ cnt,
                                                    unsigned* __restrict__ rowstart,
                                                    float* __restrict__ inv_cnt, int n) {
  __shared__ unsigned buf[1024];
  __shared__ unsigned carry_s;
  if (threadIdx.x == 0) carry_s = 0u;
  __syncthreads();
  for (int base = 0; base < n; base += 1024) {
    int i = base + (int)threadIdx.x;
    unsigned v = (i < n) ? cnt[i] : 0u;
    if (i < n) inv_cnt[i] = 1.0f / (float)(v > 1u ? v : 1u);
    buf[threadIdx.x] = v;
    __syncthreads();
    for (int off = 1; off < 1024; off <<= 1) {
      unsigned t = (threadIdx.x >= (unsigned)off) ? buf[threadIdx.x - off] : 0u;
      __syncthreads();
      buf[threadIdx.x] += t;
      __syncthreads();
    }
    unsigned carry = carry_s;
    if (i < n) rowstart[i] = carry + buf[threadIdx.x] - v;  // exclusive
    __syncthreads();
    if (threadIdx.x == 1023) carry_s = carry + buf[1023];
    __syncthreads();
  }
  if (threadIdx.x == 0) rowstart[n] = carry_s;
}

__global__ __launch_bounds__(256) void fill_kernel(const int* __restrict__ ei,
                                                   const int* __restrict__ et,
                                                   const unsigned* __restrict__ rowstart,
                                                   unsigned* __restrict__ cursor,
                                                   unsigned* __restrict__ sorted_src,
                                                   int E, int N) {
  int e = blockIdx.x * blockDim.x + threadIdx.x;
  if (e >= E) return;
  int s = et[e] * N + ei[(size_t)E + e];
  unsigned p = rowstart[s] + atomicAdd(&cursor[s], 1u);
  sorted_src[p] = (unsigned)ei[e];
}

// ---------------------------------------------------------------- mean aggregation (gather, no float atomics)
// agg[dst, 0..D) = inv_cnt[s] * sum_{p in seg s} hin[src[p], 0..D)   ; s = r*N + dst
template <int D>
__global__ __launch_bounds__(D / 4) void aggregate_kernel(const float* __restrict__ hin,
                                                          const unsigned* __restrict__ sorted_src,
                                                          const unsigned* __restrict__ rowstart,
                                                          const float* __restrict__ inv_cnt,
                                                          float* __restrict__ agg,
                                                          int r, int N) {
  const int dstn = blockIdx.x;
  const int s = r * N + dstn;
  const unsigned p0 = rowstart[s], p1 = rowstart[s + 1];
  const float inv = inv_cnt[s];
  const int c4 = (int)threadIdx.x * 4;
  float4 acc = make_float4(0.f, 0.f, 0.f, 0.f);
  for (unsigned p = p0; p < p1; ++p) {
    unsigned srcn = sorted_src[p];
    float4 t = *(const float4*)(hin + (size_t)srcn * D + c4);
    acc.x += t.x; acc.y += t.y; acc.z += t.z; acc.w += t.w;
  }
  float4 o = make_float4(acc.x * inv, acc.y * inv, acc.z * inv, acc.w * inv);
  *(float4*)(agg + (size_t)dstn * D + c4) = o;
}

// ---------------------------------------------------------------- fp32 WMMA GEMM, C += A(16-row stripe x K) * Bp(pair-packed K x H)
// blockDim = H threads = H/32 waves; wave w owns output cols [w*32, w*32+32) as two 16x16 tiles.
// A staged once in LDS (pad 4 -> conflict-free ds_load_b64); B pre-packed so each fragment is one
// global_load_b64; K fully unrolled so every access is base + immediate offset.
template <int K, int H>
__global__ __launch_bounds__(H) void gemm_acc_kernel(const float* __restrict__ A,
                                                     const float2* __restrict__ Bp,
                                                     float* __restrict__ C) {
  __shared__ float As[16][K + 4];
  const int row0 = blockIdx.x * 16;
  const int wave = threadIdx.x >> 5;
  const int lane = threadIdx.x & 31;
  const int mrow = lane & 15;     // A row / B,C column within 16-tile
  const int hsel = lane >> 4;     // 0: lanes 0-15, 1: lanes 16-31
  const int khalf = hsel << 1;    // K offset 0 or 2 (ISA f32 A/B layout)
  const int colA = wave * 32 + mrow;
  const int colB = colA + 16;

  // stage A stripe (16 x K) cooperatively with float4
  constexpr int VROW = K / 4;  // float4 vectors per row
#pragma unroll
  for (int v = threadIdx.x; v < 16 * VROW; v += H) {
    int m = v / VROW;
    int k = (v % VROW) * 4;
    *(float4*)(&As[m][k]) = *(const float4*)(A + (size_t)(row0 + m) * K + k);
  }
  __syncthreads();

  // C/D layout: VGPR v -> M = v + 8*hsel, N = lane&15
  v8f c0, c1;
#pragma unroll
  for (int v = 0; v < 8; ++v) {
    size_t rowoff = (size_t)(row0 + v + 8 * hsel) * H;
    c0[v] = C[rowoff + colA];
    c1[v] = C[rowoff + colB];
  }

  // pair index q = kk/2 + hsel  ->  rows (kk+khalf, kk+khalf+1)
  const v2f* Bq = (const v2f*)Bp + (size_t)hsel * H;
#pragma unroll
  for (int kk = 0; kk < K; kk += 4) {
    v2f a = *(const v2f*)(&As[mrow][kk + khalf]);     // ds_load_b64
    v2f b0 = Bq[(size_t)(kk >> 1) * H + colA];        // global_load_b64
    v2f b1 = Bq[(size_t)(kk >> 1) * H + colB];        // global_load_b64
    c0 = __builtin_amdgcn_wmma_f32_16x16x4_f32(false, a, false, b0, (short)0, c0, false, false);
    c1 = __builtin_amdgcn_wmma_f32_16x16x4_f32(false, a, false, b1, (short)0, c1, false, false);
  }

#pragma unroll
  for (int v = 0; v < 8; ++v) {
    size_t rowoff = (size_t)(row0 + v + 8 * hsel) * H;
    C[rowoff + colA] = c0[v];
    C[rowoff + colB] = c1[v];
  }
}

// ---------------------------------------------------------------- layer driver
// hout = b + hin@root + sum_r mean_r(hin)@W_r     (weights already pair-packed)
template <int K, int H>
static void run_layer(const float* hin, float* hout, const float2* Wp, const float2* rootp,
                      const float* b, const unsigned* ssrc, const unsigned* rowstart,
                      const float* inv_cnt, float* aggbuf, int N, int R, hipStream_t stream) {
  bias_init_kernel<<<(int)(((size_t)N * H + 255) / 256), 256, 0, stream>>>(hout, b, N, H);
  gemm_acc_kernel<K, H><<<N / 16, H, 0, stream>>>(hin, rootp, hout);
  for (int r = 0; r < R; ++r) {
    aggregate_kernel<K><<<N, K / 4, 0, stream>>>(hin, ssrc, rowstart, inv_cnt, aggbuf, r, N);
    gemm_acc_kernel<K, H><<<N / 16, H, 0, stream>>>(aggbuf, Wp + (size_t)r * (K / 2) * H, hout);
  }
}

// ---------------------------------------------------------------- host orchestration
extern "C" void kernel_launch(void* const* d_in, const int* in_sizes, int n_in,
                              void* d_out, int out_size, void* d_ws, size_t ws_size,
                              hipStream_t stream) {
  const float* x     = (const float*)d_in[0];
  const int*   ei    = (const int*)d_in[1];   // [2,E] flat: src then dst
  const int*   et    = (const int*)d_in[2];   // [E]
  const float* W1    = (const float*)d_in[3];
  const float* root1 = (const float*)d_in[4];
  const float* b1    = (const float*)d_in[5];
  const float* W2    = (const float*)d_in[6];
  const float* root2 = (const float*)d_in[7];
  const float* b2    = (const float*)d_in[8];
  const float* W3    = (const float*)d_in[9];
  const float* root3 = (const float*)d_in[10];
  const float* b3    = (const float*)d_in[11];

  const int DIN = 128, DH = 256, DOUT = 128, R = 8;
  const int N = in_sizes[0] / DIN;   // 50000
  const int E = in_sizes[2];         // 1600000
  const size_t RN = (size_t)R * N;

  // workspace carve (256B aligned)
  char* w = (char*)d_ws;
  auto carve = [&](size_t bytes) {
    void* p = (void*)w;
    w += (bytes + 255) & ~(size_t)255;
    return p;
  };
  unsigned* cnt      = (unsigned*)carve(RN * 4);
  unsigned* rowstart = (unsigned*)carve((RN + 1) * 4);
  unsigned* cursor   = (unsigned*)carve(RN * 4);
  float*    inv_cnt  = (float*)carve(RN * 4);
  unsigned* ssrc     = (unsigned*)carve((size_t)E * 4);
  float*    agg      = (float*)carve((size_t)N * DH * 4);
  float*    h1       = (float*)carve((size_t)N * DH * 4);
  float*    h2       = (float*)carve((size_t)N * DH * 4);
  float2*   W1p      = (float2*)carve((size_t)R * (DIN / 2) * DH * 8);
  float2*   r1p      = (float2*)carve((size_t)(DIN / 2) * DH * 8);
  float2*   W2p      = (float2*)carve((size_t)R * (DH / 2) * DH * 8);
  float2*   r2p      = (float2*)carve((size_t)(DH / 2) * DH * 8);
  float2*   W3p      = (float2*)carve((size_t)R * (DH / 2) * DOUT * 8);
  float2*   r3p      = (float2*)carve((size_t)(DH / 2) * DOUT * 8);
  (void)ws_size; (void)n_in; (void)out_size;

  // ---- pair-pack all weight matrices once (K even; pairs never straddle relations)
  auto pack = [&](const float* B, float2* Bpk, int Ktot, int H) {
    int npair = (Ktot / 2) * H;
    pack_pairs_kernel<<<(npair + 255) / 256, 256, 0, stream>>>(B, Bpk, Ktot, H);
  };
  pack(W1, W1p, R * DIN, DH);
  pack(root1, r1p, DIN, DH);
  pack(W2, W2p, R * DH, DH);
  pack(root2, r2p, DH, DH);
  pack(W3, W3p, R * DH, DOUT);
  pack(root3, r3p, DH, DOUT);

  // ---- CSR build over (relation, dst) segments (structure shared by all layers)
  zero_u32_kernel<<<2048, 256, 0, stream>>>(cnt, RN);
  zero_u32_kernel<<<2048, 256, 0, stream>>>(cursor, RN);
  hist_kernel<<<(E + 255) / 256, 256, 0, stream>>>(ei, et, cnt, E, N);
  scan_kernel<<<1, 1024, 0, stream>>>(cnt, rowstart, inv_cnt, (int)RN);
  fill_kernel<<<(E + 255) / 256, 256, 0, stream>>>(ei, et, rowstart, cursor, ssrc, E, N);

  // ---- three RGCN layers
  run_layer<128, 256>(x, h1, W1p, r1p, b1, ssrc, rowstart, inv_cnt, agg, N, R, stream);
  relu_kernel<<<2048, 256, 0, stream>>>(h1, (size_t)N * DH);
  run_layer<256, 256>(h1, h2, W2p, r2p, b2, ssrc, rowstart, inv_cnt, agg, N, R, stream);
  relu_kernel<<<2048, 256, 0, stream>>>(h2, (size_t)N * DH);
  run_layer<256, 128>(h2, (float*)d_out, W3p, r3p, b3, ssrc, rowstart, inv_cnt, agg, N, R, stream);
}